// ReLuTransformer_67413806678316
// MI455X (gfx1250) — compile-verified
//
#include <hip/hip_runtime.h>
#include <stdint.h>

#define D_TERMS 512
#define N_NEUR  16384

typedef float v4f __attribute__((ext_vector_type(4)));
// Exact pointee type the async-store builtin expects (per hipcc diagnostic):
typedef int v4i __attribute__((vector_size(16)));
typedef __attribute__((address_space(1))) v4i* gv4i_p;   // prints as "v4i __device__ *"
typedef __attribute__((address_space(3))) v4i* lv4i_p;   // LDS-side pointer

// --- CDNA5 async LDS->global store path (probe: fall back to NT b128 stores) ---
#if defined(__gfx1250__) && __has_builtin(__builtin_amdgcn_global_store_async_from_lds_b128)
#define USE_ASYNC_FILL 1
#else
#define USE_ASYNC_FILL 0
#endif

// K1: partial column sums of |A|. grid = (N/256, 8); each block-row handles 64 rows of A.
__global__ void k_abs_partial(const float* __restrict__ A, float* __restrict__ partial) {
    const int n = blockIdx.x * blockDim.x + threadIdx.x;   // column
    const int r = blockIdx.y;                              // D chunk 0..7
    const int CH = D_TERMS / 8;                            // 64
    const float* p = A + (size_t)r * CH * N_NEUR + n;
    float acc = 0.0f;
#pragma unroll 8
    for (int d = 0; d < CH; ++d) {
        acc += __builtin_fabsf(p[(size_t)d * N_NEUR]);
    }
    partial[(size_t)r * N_NEUR + n] = acc;
}

// K2: combine partials, do all per-neuron zonotope math.
// ws layout in:  ws[r*N + n] = partial sums, r = 0..7
// ws layout out: ws[n] = s (top scale), ws[N + n] = diag value
__global__ void k_finish(const float* __restrict__ a0, float* __restrict__ ws,
                         float* __restrict__ out_a0n) {
    const int n = blockIdx.x * blockDim.x + threadIdx.x;
    float abs_sum = 0.0f;
#pragma unroll
    for (int r = 0; r < 8; ++r) abs_sum += ws[(size_t)r * N_NEUR + n];

    const float a = a0[n];
    const float l = a - abs_sum;
    const float u = a + abs_sum;

    float slope = 0.5f * (1.0f + a / abs_sum);
    slope = fminf(fmaxf(slope, 0.0f), 1.0f);

    const float cross = (l < 0.0f && u > 0.0f) ? 1.0f : 0.0f;
    const float off   = 0.5f * fmaxf(u * (1.0f - slope), -l * slope);
    const float relu  = a > 0.0f ? a : 0.0f;
    const float a0n   = (1.0f - cross) * relu + cross * (slope * a + off);
    const float pos   = (l > 0.0f) ? 1.0f : 0.0f;

    out_a0n[n] = a0n;
    ws[n]                     = pos + cross * slope;  // s
    ws[(size_t)N_NEUR + n]    = off * cross;          // diag value
}

// K3: top = s[n] * A, float4 path, NT stores (write-once stream; A stays in L2).
__global__ void k_top(const float* __restrict__ A, const float* __restrict__ s,
                      float* __restrict__ out_top) {
    const int i = blockIdx.x * blockDim.x + threadIdx.x;   // float4 index
    const int NW4 = N_NEUR / 4;                            // 4096 float4 per row
    const int c4 = i & (NW4 - 1);
    const v4f av = ((const v4f*)A)[i];
    const v4f sv = ((const v4f*)s)[c4];
    const v4f r  = av * sv;
    __builtin_nontemporal_store(r, ((v4f*)out_top) + i);
}

// K4: zero-fill the N x N bottom block (1.07 GB). Dominant cost of the whole op.
__global__ void k_fill(float* __restrict__ base) {
    const long long Z4 = ((long long)N_NEUR * (long long)N_NEUR) / 4;  // 67,108,864
    const long long stride = (long long)gridDim.x * blockDim.x;
    long long i = (long long)blockIdx.x * blockDim.x + threadIdx.x;
#if USE_ASYNC_FILL
    __shared__ __align__(16) float zbuf[4];
    if (threadIdx.x == 0) { zbuf[0] = 0.f; zbuf[1] = 0.f; zbuf[2] = 0.f; zbuf[3] = 0.f; }
    __syncthreads();
    // Low 32 bits of a generic shared pointer are the LDS byte offset (aperture rule).
    lv4i_p lsrc = (lv4i_p)(uint32_t)(uintptr_t)&zbuf[0];
    for (; i < Z4; i += stride) {
        gv4i_p g = (gv4i_p)(uintptr_t)(base + i * 4);
        // 16B per lane, LDS -> global, tracked by ASYNCcnt; drained by S_ENDPGM.
        __builtin_amdgcn_global_store_async_from_lds_b128(g, lsrc, 0, 0);
    }
#else
    const v4f z = {0.0f, 0.0f, 0.0f, 0.0f};
    for (; i < Z4; i += stride) {
        __builtin_nontemporal_store(z, ((v4f*)base) + i);
    }
#endif
}

// K5: scatter diagonal values (runs after the fill; stream order = no race).
__global__ void k_diag(const float* __restrict__ dv, float* __restrict__ bottom) {
    const int n = blockIdx.x * blockDim.x + threadIdx.x;
    bottom[(size_t)n * N_NEUR + n] = dv[n];
}

extern "C" void kernel_launch(void* const* d_in, const int* in_sizes, int n_in,
                              void* d_out, int out_size, void* d_ws, size_t ws_size,
                              hipStream_t stream) {
    (void)in_sizes; (void)n_in; (void)out_size; (void)ws_size;
    const float* a0 = (const float*)d_in[0];
    const float* A  = (const float*)d_in[1];
    float* out = (float*)d_out;
    float* ws  = (float*)d_ws;          // needs 8 * N * 4 = 512 KB

    float* out_a0n = out;                                        // [N]
    float* out_top = out + N_NEUR;                               // [D, N]
    float* out_bot = out + N_NEUR + (size_t)D_TERMS * N_NEUR;    // [N, N]

    dim3 gp(N_NEUR / 256, 8);
    k_abs_partial<<<gp, 256, 0, stream>>>(A, ws);
    k_finish<<<N_NEUR / 256, 256, 0, stream>>>(a0, ws, out_a0n);
    k_top<<<(D_TERMS * N_NEUR / 4) / 256, 256, 0, stream>>>(A, ws, out_top);
    k_fill<<<8192, 256, 0, stream>>>(out_bot);
    k_diag<<<N_NEUR / 256, 256, 0, stream>>>(ws + N_NEUR, out_bot);
}